// GNNCritic1_8091718386003
// MI455X (gfx1250) — compile-verified
//
#include <hip/hip_runtime.h>
#include <hip/hip_bf16.h>

// ---------------------------------------------------------------------------
// GNN critic forward for MI455X (gfx1250, wave32).
// Pipeline:
//   1) deg/dinv via float atomics (self-loop included)
//   2) xw = state @ W_gcn            -- WMMA f32 16x16x4, async-LDS tiled
//   3) agg = self-loop + b_gcn init, then edge scatter (atomic fadd)
//   4) readout: x[b,c] = sum_a (relu(agg)+state) * action*10
//   5) h1 = relu(x@W1+b1), h2 = relu(h1@W2+b2)   -- same WMMA GEMM kernel
//   6) out = h2@W3 + b3 (wave-per-row dot + shuffle reduce)
// ---------------------------------------------------------------------------

typedef float v2f __attribute__((ext_vector_type(2)));
typedef float v8f __attribute__((ext_vector_type(8)));

#define TN 128          // output-column panel per block
#define KC 128          // K chunk staged in LDS
#define GEMM_BLOCK 256  // 8 waves
#define GEMM_LDS_BYTES ((KC * TN + 16 * KC) * sizeof(float))  // 64KB + 8KB

// --- CDNA5 async global->LDS DMA (GLOBAL_LOAD_ASYNC_TO_LDS_B128, ASYNCcnt) --
#if defined(__has_builtin)
#if __has_builtin(__builtin_amdgcn_global_load_async_to_lds_b128)
#define HAVE_ASYNC_LDS 1
#endif
#endif

#ifdef HAVE_ASYNC_LDS
typedef int v4i __attribute__((ext_vector_type(4)));
typedef __attribute__((address_space(1))) v4i g_v4i;   // global
typedef __attribute__((address_space(3))) v4i l_v4i;   // LDS
__device__ __forceinline__ void async_copy16(const float* gsrc, float* ldst) {
  __builtin_amdgcn_global_load_async_to_lds_b128(
      (g_v4i*)gsrc, (l_v4i*)ldst, /*offset=*/0, /*cpol=*/0);
}
#endif

__device__ __forceinline__ void wait_async0() {
#if defined(__has_builtin) && __has_builtin(__builtin_amdgcn_s_wait_asynccnt)
  __builtin_amdgcn_s_wait_asynccnt(0);
#else
  asm volatile("s_wait_asynccnt 0" ::: "memory");
#endif
}

// Generic fp32 WMMA GEMM: out[M,N] = act(A[M,K] @ W[K,N] + bias)
// Block: 16 rows x TN cols; wave w handles column tile n0 = colBase + 16*w.
__global__ void gemm_wmma_f32(const float* __restrict__ A,
                              const float* __restrict__ W,
                              const float* __restrict__ bias,
                              float* __restrict__ out,
                              int M, int K, int N, int do_relu) {
  extern __shared__ float lds[];
  float* ldsW = lds;            // [KC][TN]
  float* ldsA = lds + KC * TN;  // [16][KC]

  const int tid  = threadIdx.x;
  const int lane = tid & 31;
  const int wave = tid >> 5;
  const int half = lane >> 4;   // 0: lanes 0-15, 1: lanes 16-31
  const int l16  = lane & 15;
  const int rowBase = blockIdx.x * 16;
  const int colBase = blockIdx.y * TN;
  const int n0 = colBase + wave * 16;
  const int wcol = wave * 16;

  // Interior blocks can use the async DMA path (no bounds padding needed).
  const bool interior = (rowBase + 16 <= M) && (colBase + TN <= N) && (K % KC == 0);

  v8f acc = {};

  for (int kc = 0; kc < K; kc += KC) {
#ifdef HAVE_ASYNC_LDS
    if (interior) {
      // DMA W panel [KC x TN] straight into LDS, 16B per issue
      for (int i4 = tid; i4 < (KC * TN) / 4; i4 += GEMM_BLOCK) {
        int el = i4 * 4;
        int kr = el / TN, cc = el - kr * TN;
        async_copy16(W + (size_t)(kc + kr) * N + colBase + cc, &ldsW[el]);
      }
      // DMA A tile [16 x KC] (shared by all 8 waves)
      for (int i4 = tid; i4 < (16 * KC) / 4; i4 += GEMM_BLOCK) {
        int el = i4 * 4;
        int r = el / KC, cc = el - r * KC;
        async_copy16(A + (size_t)(rowBase + r) * K + kc + cc, &ldsA[el]);
      }
      wait_async0();
    } else
#endif
    {
      for (int idx = tid; idx < KC * TN; idx += GEMM_BLOCK) {
        int kr = idx / TN, cc = idx - kr * TN;
        int kg = kc + kr, cg = colBase + cc;
        ldsW[idx] = (kg < K && cg < N) ? W[(size_t)kg * N + cg] : 0.0f;
      }
      for (int idx = tid; idx < 16 * KC; idx += GEMM_BLOCK) {
        int r = idx / KC, cc = idx - r * KC;
        int row = rowBase + r, kg = kc + cc;
        ldsA[idx] = (row < M && kg < K) ? A[(size_t)row * K + kg] : 0.0f;
      }
    }
    __syncthreads();

    if (n0 < N) {
      // A frag (16x4 f32): lanes 0-15 rows M=0..15 hold K=k,k+1; lanes 16-31 K=k+2,k+3
      // B frag (4x16 f32): lanes 0-15 cols N hold K=k row (v0) / k+1 (v1); lanes 16-31 K=k+2/k+3
#pragma unroll
      for (int kk = 0; kk < KC; kk += 4) {
        int ka = kk + half * 2;
        v2f a = *(const v2f*)&ldsA[l16 * KC + ka];   // ds_load_b64
        v2f b;
        b.x = ldsW[ka * TN + wcol + l16];
        b.y = ldsW[(ka + 1) * TN + wcol + l16];
        acc = __builtin_amdgcn_wmma_f32_16x16x4_f32(
            /*neg_a=*/false, a, /*neg_b=*/false, b,
            /*c_mod=*/(short)0, acc, /*reuse_a=*/false, /*reuse_b=*/false);
      }
    }
    __syncthreads();
  }

  if (n0 < N) {
    int col = n0 + l16;
    float bv = (bias != nullptr && col < N) ? bias[col] : 0.0f;
#pragma unroll
    for (int r = 0; r < 8; ++r) {
      int row = rowBase + r + half * 8;  // C layout: VGPR r -> M=r (lo half) / M=8+r (hi half)
      if (row < M && col < N) {
        float v = acc[r] + bv;
        if (do_relu) v = v > 0.0f ? v : 0.0f;
        out[(size_t)row * N + col] = v;
      }
    }
  }
}

// --- degree / normalization --------------------------------------------------
__global__ void deg_init(float* dinv, int N) {
  int i = blockIdx.x * blockDim.x + threadIdx.x;
  if (i < N) dinv[i] = 1.0f;  // self-loop contributes 1
}
__global__ void deg_count(const int* __restrict__ ei, float* dinv, int E) {
  int e = blockIdx.x * blockDim.x + threadIdx.x;
  if (e < E) unsafeAtomicAdd(&dinv[ei[E + e]], 1.0f);  // dst row of edge_index
}
__global__ void deg_finalize(float* dinv, int N) {
  int i = blockIdx.x * blockDim.x + threadIdx.x;
  if (i < N) dinv[i] = rsqrtf(dinv[i]);  // deg >= 1 always (self loop)
}

// --- aggregation -------------------------------------------------------------
// agg[i,c] = xw[i,c]*dinv[i]^2 + b_gcn[c]   (self-loop term + bias)
__global__ void agg_init(const float* __restrict__ xw, const float* __restrict__ dinv,
                         const float* __restrict__ bgcn, float* __restrict__ agg,
                         int N, int C) {
  long long idx = (long long)blockIdx.x * blockDim.x + threadIdx.x;
  if (idx >= (long long)N * C) return;
  int i = (int)(idx / C), c = (int)(idx - (long long)i * C);
  float di = dinv[i];
  agg[idx] = xw[idx] * di * di + bgcn[c];
}

// one wave32 per edge; lane l covers channels 4l..4l+3 (C=128 -> exact fit)
__global__ void edge_scatter(const int* __restrict__ ei, const float* __restrict__ xw,
                             const float* __restrict__ dinv, float* __restrict__ agg,
                             int E, int C) {
  int e = (blockIdx.x * blockDim.x + threadIdx.x) >> 5;
  if (e >= E) return;
  int lane = threadIdx.x & 31;
  int s = ei[e], d = ei[E + e];
  float w = dinv[s] * dinv[d];
  const float4* xs = (const float4*)(xw + (size_t)s * C);
  float* ag = agg + (size_t)d * C;
  for (int c4 = lane; c4 * 4 < C; c4 += 32) {
    float4 v = xs[c4];
    int c = c4 * 4;
    unsafeAtomicAdd(&ag[c + 0], v.x * w);
    unsafeAtomicAdd(&ag[c + 1], v.y * w);
    unsafeAtomicAdd(&ag[c + 2], v.z * w);
    unsafeAtomicAdd(&ag[c + 3], v.w * w);
  }
}

// --- action-weighted readout -------------------------------------------------
// xr[b,c] = sum_a (relu(agg[b*A+a,c]) + state[b*A+a,c]) * action[b,a] * 10
__global__ void readout(const float* __restrict__ agg, const float* __restrict__ state,
                        const float* __restrict__ action, float* __restrict__ xr,
                        int B, int A, int C) {
  long long idx = (long long)blockIdx.x * blockDim.x + threadIdx.x;
  if (idx >= (long long)B * C) return;
  int b = (int)(idx / C), c = (int)(idx - (long long)b * C);
  float sum = 0.0f;
  for (int a = 0; a < A; ++a) {
    size_t r = (size_t)b * A + a;
    float g = agg[r * C + c];
    g = g > 0.0f ? g : 0.0f;
    g += state[r * C + c];
    sum += g * (action[r] * 10.0f);
  }
  xr[idx] = sum;
}

// --- final head: out[b] = h2[b,:] . W3 + b3 ---------------------------------
__global__ void head_dot(const float* __restrict__ h2, const float* __restrict__ W3,
                         const float* __restrict__ b3, float* __restrict__ out,
                         int B, int H) {
  int b = (blockIdx.x * blockDim.x + threadIdx.x) >> 5;
  if (b >= B) return;
  int lane = threadIdx.x & 31;
  float s = 0.0f;
  for (int j = lane; j < H; j += 32) s += h2[(size_t)b * H + j] * W3[j];
#pragma unroll
  for (int off = 16; off > 0; off >>= 1) s += __shfl_xor(s, off, 32);
  if (lane == 0) out[b] = s + b3[0];
}

// ---------------------------------------------------------------------------
extern "C" void kernel_launch(void* const* d_in, const int* in_sizes, int n_in,
                              void* d_out, int out_size, void* d_ws, size_t ws_size,
                              hipStream_t stream) {
  const float* state  = (const float*)d_in[0];
  const int*   ei     = (const int*)d_in[1];   // [2,E] (src row then dst row)
  const float* action = (const float*)d_in[2];
  const float* W_gcn  = (const float*)d_in[3];
  const float* b_gcn  = (const float*)d_in[4];
  const float* W1     = (const float*)d_in[5];
  const float* b1     = (const float*)d_in[6];
  const float* W2     = (const float*)d_in[7];
  const float* b2     = (const float*)d_in[8];
  const float* W3     = (const float*)d_in[9];
  const float* b3     = (const float*)d_in[10];
  float* out = (float*)d_out;

  const int C = in_sizes[4];            // 128
  const int H = in_sizes[6];            // 256
  const int N = in_sizes[0] / C;        // 500000
  const int E = in_sizes[1] / 2;        // 600000
  const int B = out_size;               // 50000
  const int A = in_sizes[2] / B;        // 10

  // scratch carve-out
  auto align256 = [](size_t x) { return (x + 255) & ~(size_t)255; };
  char* ws = (char*)d_ws;
  size_t o = 0;
  float* xw   = (float*)(ws + o); o = align256(o + (size_t)N * C * sizeof(float));
  float* agg  = (float*)(ws + o); o = align256(o + (size_t)N * C * sizeof(float));
  float* dinv = (float*)(ws + o); o = align256(o + (size_t)N * sizeof(float));
  float* xr   = (float*)(ws + o); o = align256(o + (size_t)B * C * sizeof(float));
  float* h1   = (float*)(ws + o); o = align256(o + (size_t)B * H * sizeof(float));
  float* h2   = (float*)(ws + o); o = align256(o + (size_t)B * H * sizeof(float));
  (void)ws_size; (void)n_in;

  // 1) degrees -> dinv
  deg_init<<<(N + 255) / 256, 256, 0, stream>>>(dinv, N);
  deg_count<<<(E + 255) / 256, 256, 0, stream>>>(ei, dinv, E);
  deg_finalize<<<(N + 255) / 256, 256, 0, stream>>>(dinv, N);

  // 2) xw = state @ W_gcn   [N,C]
  {
    dim3 grid((N + 15) / 16, (C + TN - 1) / TN);
    gemm_wmma_f32<<<grid, GEMM_BLOCK, GEMM_LDS_BYTES, stream>>>(
        state, W_gcn, nullptr, xw, N, C, C, 0);
  }

  // 3) aggregation
  {
    long long total = (long long)N * C;
    agg_init<<<(unsigned)((total + 255) / 256), 256, 0, stream>>>(xw, dinv, b_gcn, agg, N, C);
    edge_scatter<<<(E + 7) / 8, 256, 0, stream>>>(ei, xw, dinv, agg, E, C);
  }

  // 4) readout -> xr [B,C]
  {
    long long total = (long long)B * C;
    readout<<<(unsigned)((total + 255) / 256), 256, 0, stream>>>(agg, state, action, xr, B, A, C);
  }

  // 5) MLP layers (WMMA)
  {
    dim3 grid1((B + 15) / 16, (H + TN - 1) / TN);
    gemm_wmma_f32<<<grid1, GEMM_BLOCK, GEMM_LDS_BYTES, stream>>>(xr, W1, b1, h1, B, C, H, 1);
    dim3 grid2((B + 15) / 16, (H + TN - 1) / TN);
    gemm_wmma_f32<<<grid2, GEMM_BLOCK, GEMM_LDS_BYTES, stream>>>(h1, W2, b2, h2, B, H, H, 1);
  }

  // 6) head
  head_dot<<<(B + 7) / 8, 256, 0, stream>>>(h2, W3, b3, out, B, H);
}